// GraphAttentionLayer_16784732193026
// MI455X (gfx1250) — compile-verified
//
#include <hip/hip_runtime.h>
#include <cstddef>

// GAT forward: B=8, N=2048, K=256 (in), D=64 (out), alpha=0.2, ELU(alpha=1)
#define B_ 8
#define N_ 2048
#define K_ 256
#define D_ 64
#define ALPHA_ 0.2f

typedef __attribute__((ext_vector_type(2))) float v2f;
typedef __attribute__((ext_vector_type(8))) float v8f;

__device__ __forceinline__ float lrelu(float x) { return x > 0.f ? x : ALPHA_ * x; }

// D = A(16x4 f32) * B(4x16 f32) + C(16x16 f32), wave32 matrix pipe.
// Per-lane fragments (ISA 7.12.2): lane l, m/n = l&15, kk = 2*(l>>4);
//   A: {A[m][k0+kk], A[m][k0+kk+1]}   B: {B[k0+kk][n], B[k0+kk+1][n]}
//   C/D: vgpr v, lanes 0-15 -> (M=v, N=l), lanes 16-31 -> (M=v+8, N=l-16)
__device__ __forceinline__ v8f wmma4(v2f a, v2f b, v8f c) {
  return __builtin_amdgcn_wmma_f32_16x16x4_f32(false, a, false, b, (short)0, c,
                                               false, false);
}

// ---------------------------------------------------------------------------
// Kernel 1: Wh = h @ W   (rows flat over b*N; 4 waves x 16 rows = 64 rows/block)
// W staged transposed in LDS: sWT[d*132 + k_local], 128-K phases, 33.8 KB LDS.
// Stride 132 (== 4 mod 64, even): conflict-free 8B fragment loads across wave32.
// ---------------------------------------------------------------------------
__global__ __launch_bounds__(128) void k_wh(const float* __restrict__ h,
                                            const float* __restrict__ W,
                                            float* __restrict__ Wh) {
  __shared__ float sWT[64 * 132];
  const int tid = threadIdx.x;
  const int wave = tid >> 5, lane = tid & 31;
  const int m = lane & 15, hk = lane >> 4, kk = hk * 2;
  const int rowBase = blockIdx.x * 64 + wave * 16;
  const float* hrow = h + (size_t)(rowBase + m) * K_;

  v8f c0 = {}, c1 = {}, c2 = {}, c3 = {};
  for (int p = 0; p < 2; ++p) {  // two 128-wide K phases
    __syncthreads();
    for (int idx = tid; idx < 128 * 64; idx += 128) {
      int kl = idx >> 6, d = idx & 63;
      sWT[d * 132 + kl] = W[(p * 128 + kl) * D_ + d];
    }
    __syncthreads();
#pragma unroll 8
    for (int k0 = 0; k0 < 128; k0 += 4) {
      v2f a = *(const v2f*)(hrow + p * 128 + k0 + kk);
      v2f b0 = *(const v2f*)&sWT[(0 * 16 + m) * 132 + k0 + kk];
      v2f b1 = *(const v2f*)&sWT[(1 * 16 + m) * 132 + k0 + kk];
      v2f b2 = *(const v2f*)&sWT[(2 * 16 + m) * 132 + k0 + kk];
      v2f b3 = *(const v2f*)&sWT[(3 * 16 + m) * 132 + k0 + kk];
      c0 = wmma4(a, b0, c0);
      c1 = wmma4(a, b1, c1);
      c2 = wmma4(a, b2, c2);
      c3 = wmma4(a, b3, c3);
    }
  }
#pragma unroll
  for (int v = 0; v < 8; ++v) {
    int row = rowBase + v + 8 * hk;
    float* o = Wh + (size_t)row * D_ + m;
    o[0] = c0[v];
    o[16] = c1[v];
    o[32] = c2[v];
    o[48] = c3[v];
  }
}

// ---------------------------------------------------------------------------
// Kernel 2: f1[r] = Wh[r,:] . a1 ;  f2[r] = Wh[r,:] . a2
// ---------------------------------------------------------------------------
__global__ __launch_bounds__(256) void k_f12(const float* __restrict__ Wh,
                                             const float* __restrict__ a,
                                             float* __restrict__ f1,
                                             float* __restrict__ f2) {
  const int row = blockIdx.x * 256 + threadIdx.x;
  const float4* whr = (const float4*)(Wh + (size_t)row * D_);
  const float4* a1 = (const float4*)a;
  const float4* a2 = (const float4*)(a + D_);
  float s1 = 0.f, s2 = 0.f;
#pragma unroll
  for (int q = 0; q < D_ / 4; ++q) {
    float4 v = whr[q], x = a1[q], y = a2[q];
    s1 += v.x * x.x + v.y * x.y + v.z * x.z + v.w * x.w;
    s2 += v.x * y.x + v.y * y.y + v.z * y.z + v.w * y.w;
  }
  f1[row] = s1;
  f2[row] = s2;
}

// ---------------------------------------------------------------------------
// Kernel 3: mx[b] = max_i f1[b,i]  (softmax-over-i max factors through lrelu)
// ---------------------------------------------------------------------------
__global__ __launch_bounds__(256) void k_rowmax(const float* __restrict__ f1,
                                                float* __restrict__ mx) {
  __shared__ float red[256];
  const int b = blockIdx.x;
  float v = -__builtin_inff();
  for (int i = threadIdx.x; i < N_; i += 256) v = fmaxf(v, f1[b * N_ + i]);
  red[threadIdx.x] = v;
  __syncthreads();
  for (int s = 128; s > 0; s >>= 1) {
    if (threadIdx.x < s)
      red[threadIdx.x] = fmaxf(red[threadIdx.x], red[threadIdx.x + s]);
    __syncthreads();
  }
  if (threadIdx.x == 0) mx[b] = red[0];
}

// ---------------------------------------------------------------------------
// Kernel 4: per (b,j): m[j] = lrelu(mx+f2[j]); w[j] = 1 / sum_i exp(e[i,j]-m[j])
// ---------------------------------------------------------------------------
__global__ __launch_bounds__(256) void k_denom(const float* __restrict__ f1,
                                               const float* __restrict__ f2,
                                               const float* __restrict__ mx,
                                               float* __restrict__ mj,
                                               float* __restrict__ wj) {
  __shared__ float sf1[N_];
  const int b = blockIdx.x >> 3, jc = blockIdx.x & 7;
  for (int i = threadIdx.x; i < N_; i += 256) sf1[i] = f1[b * N_ + i];
  __syncthreads();
  const int j = b * N_ + jc * 256 + threadIdx.x;
  const float f2j = f2[j];
  const float mv = lrelu(mx[b] + f2j);
  float s = 0.f;
#pragma unroll 8
  for (int i = 0; i < N_; ++i) s += __expf(lrelu(sf1[i] + f2j) - mv);
  mj[j] = mv;
  wj[j] = 1.0f / s;
}

// ---------------------------------------------------------------------------
// Kernel 5: h'[b,i,:] = sum_j P[i,j] * Wh[b,j,:], P computed in-register
// (flash-style, never materializing P), then fused ELU.
// Block = 8 waves x 16 rows = 128 i-rows; Wh chunk (128 j) staged transposed.
// ---------------------------------------------------------------------------
__global__ __launch_bounds__(256) void k_attn_out(
    const float* __restrict__ Wh, const float* __restrict__ f1,
    const float* __restrict__ f2, const float* __restrict__ mj,
    const float* __restrict__ wj, float* __restrict__ out) {
  __shared__ float sWhT[64 * 132];  // WhT[d*132 + j_local]
  __shared__ float sf2[128], smj[128], swj[128];
  const int tid = threadIdx.x;
  const int wave = tid >> 5, lane = tid & 31;
  const int m = lane & 15, hk = lane >> 4, kk = hk * 2;
  const int b = blockIdx.x >> 4, ic = blockIdx.x & 15;
  const int iBase = ic * 128 + wave * 16;
  const float f1m = f1[b * N_ + iBase + m];

  v8f c0 = {}, c1 = {}, c2 = {}, c3 = {};
  for (int jc = 0; jc < N_ / 128; ++jc) {
    __syncthreads();
    for (int idx = tid; idx < 128 * 64; idx += 256) {
      int jj = idx >> 6, d = idx & 63;
      sWhT[d * 132 + jj] = Wh[(size_t)(b * N_ + jc * 128 + jj) * D_ + d];
    }
    if (tid < 128) {
      int jg = b * N_ + jc * 128 + tid;
      sf2[tid] = f2[jg];
      smj[tid] = mj[jg];
      swj[tid] = wj[jg];
    }
    __syncthreads();
#pragma unroll 4
    for (int j0 = 0; j0 < 128; j0 += 4) {
      const int ja = j0 + kk;
      float e0 = __expf(lrelu(f1m + sf2[ja]) - smj[ja]) * swj[ja];
      float e1 = __expf(lrelu(f1m + sf2[ja + 1]) - smj[ja + 1]) * swj[ja + 1];
      v2f a;
      a.x = e0;
      a.y = e1;
      v2f b0 = *(const v2f*)&sWhT[(0 * 16 + m) * 132 + ja];
      v2f b1 = *(const v2f*)&sWhT[(1 * 16 + m) * 132 + ja];
      v2f b2 = *(const v2f*)&sWhT[(2 * 16 + m) * 132 + ja];
      v2f b3 = *(const v2f*)&sWhT[(3 * 16 + m) * 132 + ja];
      c0 = wmma4(a, b0, c0);
      c1 = wmma4(a, b1, c1);
      c2 = wmma4(a, b2, c2);
      c3 = wmma4(a, b3, c3);
    }
  }
#pragma unroll
  for (int v = 0; v < 8; ++v) {
    int i = iBase + v + 8 * hk;
    float* o = out + (size_t)(b * N_ + i) * D_ + m;
    float x0 = c0[v], x1 = c1[v], x2 = c2[v], x3 = c3[v];
    o[0] = x0 > 0.f ? x0 : expm1f(x0);
    o[16] = x1 > 0.f ? x1 : expm1f(x1);
    o[32] = x2 > 0.f ? x2 : expm1f(x2);
    o[48] = x3 > 0.f ? x3 : expm1f(x3);
  }
}

// ---------------------------------------------------------------------------
extern "C" void kernel_launch(void* const* d_in, const int* in_sizes, int n_in,
                              void* d_out, int out_size, void* d_ws,
                              size_t ws_size, hipStream_t stream) {
  (void)in_sizes;
  (void)n_in;
  (void)out_size;
  (void)ws_size;
  const float* h = (const float*)d_in[0];  // (8, 2048, 256)
  const float* W = (const float*)d_in[1];  // (256, 64)
  const float* a = (const float*)d_in[2];  // (128, 1)
  float* out = (float*)d_out;              // (8, 2048, 64)

  // Workspace layout (floats): Wh | f1 | f2 | mj | wj | mx  (~4.25 MB total)
  float* Wh = (float*)d_ws;
  float* f1 = Wh + (size_t)B_ * N_ * D_;
  float* f2 = f1 + (size_t)B_ * N_;
  float* mj = f2 + (size_t)B_ * N_;
  float* wj = mj + (size_t)B_ * N_;
  float* mx = wj + (size_t)B_ * N_;

  k_wh<<<(B_ * N_) / 64, 128, 0, stream>>>(h, W, Wh);
  k_f12<<<(B_ * N_) / 256, 256, 0, stream>>>(Wh, a, f1, f2);
  k_rowmax<<<B_, 256, 0, stream>>>(f1, mx);
  k_denom<<<B_ * (N_ / 256), 256, 0, stream>>>(f1, f2, mx, mj, wj);
  k_attn_out<<<B_ * (N_ / 128), 256, 0, stream>>>(Wh, f1, f2, mj, wj, out);
}